// KVCache_6390911337260
// MI455X (gfx1250) — compile-verified
//
#include <hip/hip_runtime.h>
#include <cstdint>

// Problem shape (from the reference): B,L,H,D,Q
#define THREADS 256
#define CHUNKS  4

constexpr int    B = 8, L = 4096, H = 16, D = 64, Q = 1024;
constexpr size_t CACHE_ELEMS = (size_t)B * L * H * D;   // 33,554,432 floats (128 MiB)
constexpr size_t CACHE_F4    = CACHE_ELEMS / 4;         // 8,388,608 float4
constexpr size_t HD_F4       = (size_t)H * D / 4;       // 256 float4 per (b,q) row

// ---------------------------------------------------------------------------
// Kernel 1: bulk copy k_cache||v_cache -> d_out via the CDNA5 async LDS path.
// Each thread: 4 async 16B global->LDS loads, wait ASYNCcnt, 4 async LDS->global
// stores. NT temporal hints: 256 MiB streamed once, larger than the 192 MB L2.
// ---------------------------------------------------------------------------
__global__ __launch_bounds__(THREADS)
void kv_copy_async(const float4* __restrict__ kc,
                   const float4* __restrict__ vc,
                   float4* __restrict__ out) {
  __shared__ __align__(16) unsigned char stage[THREADS * CHUNKS * 16];  // 16 KiB

  const uint32_t tid  = threadIdx.x;
  const size_t   base = (size_t)blockIdx.x * (THREADS * CHUNKS);

  // Concatenated f4 space: [0, CACHE_F4) = k_cache, [CACHE_F4, 2*CACHE_F4) = v_cache.
  // Block span (1024 f4) divides CACHE_F4, so a block never straddles the halves.
  const float4* src = (base < CACHE_F4) ? (kc + base) : (vc + (base - CACHE_F4));
  float4*       dst = out + base;

  // Generic pointer to LDS: bits [31:0] are the LDS byte offset (ISA aperture rule).
  const uint32_t lds0 = (uint32_t)(uintptr_t)(&stage[0]);

#pragma unroll
  for (int k = 0; k < CHUNKS; ++k) {
    const uint32_t lds = lds0 + (uint32_t)(k * THREADS + tid) * 16u;
    const float4*  g   = src + (size_t)k * THREADS + tid;
    asm volatile("global_load_async_to_lds_b128 %0, %1, off th:TH_LOAD_NT"
                 :: "v"(lds), "v"(g) : "memory");
  }

  // Wave-local: all this wave's async loads have deposited into LDS.
  asm volatile("s_wait_asynccnt 0x0" ::: "memory");

#pragma unroll
  for (int k = 0; k < CHUNKS; ++k) {
    const uint32_t lds = lds0 + (uint32_t)(k * THREADS + tid) * 16u;
    float4*        g   = dst + (size_t)k * THREADS + tid;
    asm volatile("global_store_async_from_lds_b128 %0, %1, off th:TH_STORE_NT"
                 :: "v"(g), "v"(lds) : "memory");
  }
  // s_endpgm performs an implicit wait-idle: async stores drain before wave exit.
}

// ---------------------------------------------------------------------------
// Kernel 2: scatter the new (B,Q,H,D) values into d_out at cache slot pos-1.
// One block per (b,q); each thread moves one float4 of the 4 KiB row for K and V.
// ---------------------------------------------------------------------------
__global__ __launch_bounds__(THREADS)
void kv_scatter(const float4* __restrict__ kv,
                const float4* __restrict__ vv,
                const int* __restrict__ pos,
                float4* __restrict__ out) {
  const int gid = blockIdx.x;        // b*Q + q
  const int b   = gid >> 10;         // Q == 1024
  const int idx = pos[gid] - 1;      // 1-indexed positions in the reference
  const int t   = threadIdx.x;       // one float4 of the H*D row

  const size_t srcOff = (size_t)gid * HD_F4 + t;
  const size_t dstOff = ((size_t)b * L + idx) * HD_F4 + t;

  out[dstOff]            = kv[srcOff];   // k_out half
  out[CACHE_F4 + dstOff] = vv[srcOff];   // v_out half
}

// ---------------------------------------------------------------------------
// Harness entry. Inputs (setup_inputs order): k_cache, v_cache, k_val, v_val,
// input_pos. d_out = k_out || v_out, all float32.
// ---------------------------------------------------------------------------
extern "C" void kernel_launch(void* const* d_in, const int* in_sizes, int n_in,
                              void* d_out, int out_size, void* d_ws, size_t ws_size,
                              hipStream_t stream) {
  const float4* kc  = (const float4*)d_in[0];
  const float4* vc  = (const float4*)d_in[1];
  const float4* kv  = (const float4*)d_in[2];
  const float4* vv  = (const float4*)d_in[3];
  const int*    pos = (const int*)d_in[4];
  float4*       out = (float4*)d_out;

  const int copyBlocks = (int)((2 * CACHE_F4) / (THREADS * CHUNKS));  // 16384
  kv_copy_async<<<copyBlocks, THREADS, 0, stream>>>(kc, vc, out);
  kv_scatter<<<B * Q, THREADS, 0, stream>>>(kv, vv, pos, out);        // 8192 blocks
}